// CRDLoss_69853348102538
// MI455X (gfx1250) — compile-verified
//
#include <hip/hip_runtime.h>
#include <math.h>

#define NDATA   50000
#define FEAT    128
#define SDIM    512
#define TDIM    2048
#define BATCH   64
#define KTOT    16385              // NCE_K + 1
#define INV_T   (1.0f / 0.07f)
#define MPN     0.32768f           // m * Pn = 16384 / 50000
#define EPSV    1e-7f

// workspace layout (float offsets)
#define WS_V1    0
#define WS_V2    8192
#define WS_GA    16384                   // mem_v1 . v2  (50000 x 64)
#define WS_GB    (16384 + 3200000)       // mem_v2 . v1  (50000 x 64)
#define WS_ZPART (16384 + 6400000)       // 128 floats
#define WS_Z     (WS_ZPART + 128)        // 2 floats
#define WS_LPART (WS_Z + 2)              // 128 floats

typedef __attribute__((ext_vector_type(2))) float v2f;
typedef __attribute__((ext_vector_type(8))) float v8f;

// ---------------------------------------------------------------------------
// 1) Embed: v = normalize(x @ W^T + b), one block per (batch row, view)
// ---------------------------------------------------------------------------
__global__ void embed_kernel(const float* __restrict__ f_s, const float* __restrict__ f_t,
                             const float* __restrict__ Ws, const float* __restrict__ bs,
                             const float* __restrict__ Wt, const float* __restrict__ bt,
                             float* __restrict__ ws) {
    __shared__ float xs[TDIM];
    __shared__ float red[FEAT];
    const int b = blockIdx.x;
    const int j = threadIdx.x;                 // 0..127, output feature
    const bool tpath = (blockIdx.y == 1);
    const int xdim = tpath ? TDIM : SDIM;
    const float* x    = (tpath ? f_t : f_s) + (size_t)b * xdim;
    const float* W    = tpath ? Wt : Ws;
    const float* bias = tpath ? bt : bs;
    float* vout = ws + (tpath ? WS_V2 : WS_V1);

    for (int i = j; i < xdim; i += FEAT) xs[i] = x[i];
    __syncthreads();

    const float* wrow = W + (size_t)j * xdim;
    float h = bias[j];
    for (int k = 0; k < xdim; ++k) h += xs[k] * wrow[k];

    red[j] = h * h;
    __syncthreads();
    for (int s = 64; s > 0; s >>= 1) {
        if (j < s) red[j] += red[j + s];
        __syncthreads();
    }
    vout[b * FEAT + j] = h / sqrtf(red[0]);
}

// ---------------------------------------------------------------------------
// 2) Dense GEMM G[r][b] = dot(mem[r], v[b]) via V_WMMA_F32_16X16X4_F32.
//    One wave computes a 16(row) x 64(batch) tile, K=128 in steps of 4.
// ---------------------------------------------------------------------------
__global__ void __launch_bounds__(256) gemm_kernel(const float* __restrict__ mem_v1,
                                                   const float* __restrict__ mem_v2,
                                                   float* __restrict__ ws) {
    __shared__ float vlds[64 * 130];           // padded stride 130 (bank-conflict-free)
    const int tid = threadIdx.x;
    const bool second = (blockIdx.y == 1);     // 0: mem_v1 x v2 -> GA ; 1: mem_v2 x v1 -> GB
    const float* vsrc = ws + (second ? WS_V1 : WS_V2);
    const float* mem  = second ? mem_v2 : mem_v1;
    float* G = ws + (second ? WS_GB : WS_GA);

    for (int i = tid; i < 64 * FEAT; i += 256)
        vlds[(i >> 7) * 130 + (i & 127)] = vsrc[i];
    __syncthreads();

    const int wave = tid >> 5;
    const int lane = tid & 31;
    int rowTile = blockIdx.x * 8 + wave;
    if (rowTile > 3124) rowTile = 3124;        // wave-uniform clamp: EXEC stays all-ones
    const int r0   = rowTile * 16;
    const int half = lane >> 4;                // 0/1 : which K-pair this lane carries
    const int m16  = lane & 15;

    const float* arow  = mem + (size_t)(r0 + m16) * FEAT + 2 * half;
    const float* bbase = &vlds[m16 * 130 + 2 * half];

    v8f acc0 = {}, acc1 = {}, acc2 = {}, acc3 = {};
#pragma unroll 8
    for (int kk = 0; kk < FEAT; kk += 4) {
        v2f a  = *(const v2f*)(arow + kk);                   // A: rows r0+m, K pair
        v2f b0 = *(const v2f*)(bbase + kk);                  // B: cols  0..15
        v2f b1 = *(const v2f*)(bbase + 16 * 130 + kk);       // B: cols 16..31
        v2f b2 = *(const v2f*)(bbase + 32 * 130 + kk);       // B: cols 32..47
        v2f b3 = *(const v2f*)(bbase + 48 * 130 + kk);       // B: cols 48..63
        acc0 = __builtin_amdgcn_wmma_f32_16x16x4_f32(false, a, false, b0, (short)0, acc0, false, false);
        acc1 = __builtin_amdgcn_wmma_f32_16x16x4_f32(false, a, false, b1, (short)0, acc1, false, false);
        acc2 = __builtin_amdgcn_wmma_f32_16x16x4_f32(false, a, false, b2, (short)0, acc2, false, false);
        acc3 = __builtin_amdgcn_wmma_f32_16x16x4_f32(false, a, false, b3, (short)0, acc3, false, false);
    }

    // C/D layout: VGPR r -> M = r + 8*half, N = lane&15
    const int mrow = r0 + half * 8;
    float* gout = G + (size_t)mrow * 64 + m16;
#pragma unroll
    for (int r = 0; r < 8; ++r) {
        gout[(size_t)r * 64 + 0]  = acc0[r];
        gout[(size_t)r * 64 + 16] = acc1[r];
        gout[(size_t)r * 64 + 32] = acc2[r];
        gout[(size_t)r * 64 + 48] = acc3[r];
    }
}

// ---------------------------------------------------------------------------
// 3) Z partial sums: per-batch sums of exp(G[ci]/T)  (deterministic tree red.)
// ---------------------------------------------------------------------------
__global__ void zsum_kernel(const float* __restrict__ ws, const int* __restrict__ idx,
                            const int* __restrict__ cidx, float* __restrict__ wsw) {
    __shared__ float r1[256], r2[256];
    const int b = blockIdx.x;
    const int tid = threadIdx.x;
    const float* Ga = ws + WS_GA;
    const float* Gb = ws + WS_GB;
    float s1 = 0.f, s2 = 0.f;
    for (int k = tid; k < KTOT; k += 256) {
        int r = (k == 0) ? idx[b] : cidx[b * KTOT + k];
        s1 += expf(Gb[(size_t)r * 64 + b] * INV_T);   // out_v1 terms
        s2 += expf(Ga[(size_t)r * 64 + b] * INV_T);   // out_v2 terms
    }
    r1[tid] = s1; r2[tid] = s2;
    __syncthreads();
    for (int s = 128; s > 0; s >>= 1) {
        if (tid < s) { r1[tid] += r1[tid + s]; r2[tid] += r2[tid + s]; }
        __syncthreads();
    }
    if (tid == 0) {
        wsw[WS_ZPART + b]      = r1[0];
        wsw[WS_ZPART + 64 + b] = r2[0];
    }
}

__global__ void zfinal_kernel(float* __restrict__ ws) {
    if (threadIdx.x == 0) {
        const float* zp = ws + WS_ZPART;
        float t1 = 0.f, t2 = 0.f;
        for (int i = 0; i < 64; ++i) { t1 += zp[i]; t2 += zp[64 + i]; }
        ws[WS_Z + 0] = t1 / (64.0f * (float)KTOT) * (float)NDATA;   // Z_v1
        ws[WS_Z + 1] = t2 / (64.0f * (float)KTOT) * (float)NDATA;   // Z_v2
    }
}

// ---------------------------------------------------------------------------
// 4) Loss partial sums with known Z
// ---------------------------------------------------------------------------
__global__ void loss_kernel(const float* __restrict__ ws, const int* __restrict__ idx,
                            const int* __restrict__ cidx, float* __restrict__ wsw) {
    __shared__ float r1[256], r2[256];
    const int b = blockIdx.x;
    const int tid = threadIdx.x;
    const float* Ga = ws + WS_GA;
    const float* Gb = ws + WS_GB;
    const float Z1 = ws[WS_Z + 0];
    const float Z2 = ws[WS_Z + 1];
    const float c = MPN + EPSV;
    float s1 = 0.f, s2 = 0.f;
    for (int k = tid; k < KTOT; k += 256) {
        int r = (k == 0) ? idx[b] : cidx[b * KTOT + k];
        float e1 = expf(Gb[(size_t)r * 64 + b] * INV_T) / Z1;
        float e2 = expf(Ga[(size_t)r * 64 + b] * INV_T) / Z2;
        if (k == 0) {
            s1 += logf(e1 / (e1 + c));
            s2 += logf(e2 / (e2 + c));
        } else {
            s1 += logf(MPN / (e1 + c));
            s2 += logf(MPN / (e2 + c));
        }
    }
    r1[tid] = s1; r2[tid] = s2;
    __syncthreads();
    for (int s = 128; s > 0; s >>= 1) {
        if (tid < s) { r1[tid] += r1[tid + s]; r2[tid] += r2[tid + s]; }
        __syncthreads();
    }
    if (tid == 0) {
        wsw[WS_LPART + b]      = r1[0];
        wsw[WS_LPART + 64 + b] = r2[0];
    }
}

__global__ void lossfinal_kernel(const float* __restrict__ ws, float* __restrict__ out) {
    if (threadIdx.x == 0) {
        const float* lp = ws + WS_LPART;
        float t1 = 0.f, t2 = 0.f;
        for (int i = 0; i < 64; ++i) { t1 += lp[i]; t2 += lp[64 + i]; }
        out[0] = -(t1 / (float)BATCH) - (t2 / (float)BATCH);   // s_loss + t_loss
    }
}

// ---------------------------------------------------------------------------
// 5) Copy memory banks to output (new_mem before row updates)
// ---------------------------------------------------------------------------
__global__ void copy_kernel(const float* __restrict__ m1, const float* __restrict__ m2,
                            float* __restrict__ out) {
    const size_t n = (size_t)NDATA * FEAT;
    size_t i = (size_t)blockIdx.x * 256 + threadIdx.x;
    if (i < n) {
        if (blockIdx.y == 0) out[1 + i]     = m1[i];
        else                 out[1 + n + i] = m2[i];
    }
}

// ---------------------------------------------------------------------------
// 6) Row updates: pos = 0.5*mem[idx[b]] + 0.5*v[b], renormalize, scatter.
//    Sequential over b in one block -> deterministic "last index wins".
// ---------------------------------------------------------------------------
__global__ void update_kernel(const float* __restrict__ m1, const float* __restrict__ m2,
                              const int* __restrict__ idx, const float* __restrict__ ws,
                              float* __restrict__ out) {
    __shared__ float red[FEAT];
    const int j = threadIdx.x;
    const size_t n = (size_t)NDATA * FEAT;
    for (int t = 0; t < 2; ++t) {
        const float* mem = t ? m2 : m1;
        const float* v   = ws + (t ? WS_V2 : WS_V1);
        float* o = out + 1 + (t ? n : 0);
        for (int b = 0; b < BATCH; ++b) {
            int row = idx[b];
            float p = mem[(size_t)row * FEAT + j] * 0.5f + v[b * FEAT + j] * 0.5f;
            red[j] = p * p;
            __syncthreads();
            for (int s = 64; s > 0; s >>= 1) {
                if (j < s) red[j] += red[j + s];
                __syncthreads();
            }
            o[(size_t)row * FEAT + j] = p / sqrtf(red[0]);
            __syncthreads();
        }
    }
}

// ---------------------------------------------------------------------------
extern "C" void kernel_launch(void* const* d_in, const int* in_sizes, int n_in,
                              void* d_out, int out_size, void* d_ws, size_t ws_size,
                              hipStream_t stream) {
    const float* f_s = (const float*)d_in[0];
    const float* f_t = (const float*)d_in[1];
    const float* Ws  = (const float*)d_in[2];
    const float* bs  = (const float*)d_in[3];
    const float* Wt  = (const float*)d_in[4];
    const float* bt  = (const float*)d_in[5];
    const float* m1  = (const float*)d_in[6];
    const float* m2  = (const float*)d_in[7];
    const int* idx   = (const int*)d_in[8];
    const int* cidx  = (const int*)d_in[9];
    float* out = (float*)d_out;
    float* ws  = (float*)d_ws;

    embed_kernel<<<dim3(BATCH, 2), 128, 0, stream>>>(f_s, f_t, Ws, bs, Wt, bt, ws);
    gemm_kernel<<<dim3(391, 2), 256, 0, stream>>>(m1, m2, ws);   // 391*8 = 3128 >= 3125 row tiles
    zsum_kernel<<<BATCH, 256, 0, stream>>>(ws, idx, cidx, ws);
    zfinal_kernel<<<1, 64, 0, stream>>>(ws);
    loss_kernel<<<BATCH, 256, 0, stream>>>(ws, idx, cidx, ws);
    lossfinal_kernel<<<1, 64, 0, stream>>>(ws, out);
    copy_kernel<<<dim3(25000, 2), 256, 0, stream>>>(m1, m2, out);
    update_kernel<<<1, FEAT, 0, stream>>>(m1, m2, idx, ws, out);
}